// GetCostVolume_50946902065807
// MI455X (gfx1250) — compile-verified
//
#include <hip/hip_runtime.h>
#include <cstdint>

typedef float v4f __attribute__((ext_vector_type(4)));

namespace {
constexpr int C  = 16;
constexpr int H  = 256;
constexpr int W  = 512;
constexpr int ND = 32;
constexpr float BASE = 0.24f;
constexpr float PI_F = 3.14159265358979323846f;
}

// ---------------------------------------------------------------------------
// Kernel 1: out[0, c, d, h, w] = x[0, c, h, w]  (channels 0..15, all 32 disps)
// One thread per (c,h,w/4); b128 read once, 32 non-temporal b128 stores.
// ---------------------------------------------------------------------------
__global__ __launch_bounds__(256) void cv_ref_broadcast(
    const float* __restrict__ x, float* __restrict__ out) {
  const int t  = blockIdx.x * 256 + threadIdx.x;   // 16*256*128 threads total
  const int w4 = t & (W / 4 - 1);
  const int h  = (t >> 7) & (H - 1);
  const int c  = t >> 15;                          // 0..15
  const v4f val = *((const v4f*)(x + ((size_t)c * H + h) * (size_t)W) + w4);
  float* o = out + (((size_t)c * ND) * H + h) * (size_t)W + (size_t)w4 * 4;
#pragma unroll
  for (int d = 0; d < ND; ++d)
    __builtin_nontemporal_store(val, (v4f*)(o + (size_t)d * (H * (size_t)W)));
}

// ---------------------------------------------------------------------------
// Kernel 2: spherical vertical warp (channels 16..31).
// One thread per (d,h,w/4). Depth staged via CDNA5 async global->LDS copy
// (ASYNCcnt), bilinear zero-padded gather from L2-resident y, NT b128 stores.
// ---------------------------------------------------------------------------
__global__ __launch_bounds__(256) void cv_warp(
    const float* __restrict__ y, const float* __restrict__ depth,
    float* __restrict__ out) {
  __shared__ v4f sdepth[256];
  const int t  = blockIdx.x * 256 + threadIdx.x;   // 32*256*128 threads total
  const int w4 = t & (W / 4 - 1);
  const int h  = (t >> 7) & (H - 1);
  const int d  = t >> 15;                          // 0..31
  const int w  = w4 * 4;

  // --- CDNA5 async tensor-path: per-lane 16B global->LDS copy of depth ---
  const float* gdep = depth + (((size_t)d * H + h) * (size_t)W + w);
  const unsigned ldsOff = (unsigned)(uintptr_t)&sdepth[threadIdx.x];
  asm volatile("global_load_async_to_lds_b128 %0, %1, off"
               :: "v"(ldsOff), "v"((unsigned long long)(uintptr_t)gdep)
               : "memory");

  // overlap grid math with the async copy
  const float theta = ((float)h + 0.5f) * (PI_F / (float)H) - 0.5f * PI_F;
  const float st = sinf(theta), ct = cosf(theta);
  const float hScale = (float)H / ((float)H - 1.0f);
  const float wScale = (float)W / ((float)W - 1.0f);

#if __has_builtin(__builtin_amdgcn_s_wait_asynccnt)
  __builtin_amdgcn_s_wait_asynccnt(0);
#else
  asm volatile("s_wait_asynccnt 0x0" ::: "memory");
#endif
  const v4f dep = sdepth[threadIdx.x];

  int   off00[4], off01[4], off10[4], off11[4];
  float q00[4], q01[4], q10[4], q11[4];
#pragma unroll
  for (int j = 0; j < 4; ++j) {
    const float dp  = dep[j];
    const float dv  = (atanf((dp * st + BASE) / (dp * ct)) - theta)
                      * ((float)H / PI_F);
    const float ypx = __builtin_isfinite(dv) ? ((float)h + dv) : 0.0f;
    const float iy  = ypx * hScale - 0.5f;                 // align_corners=False
    const float ix  = (float)(w + j) * wScale - 0.5f;

    const float fx0 = floorf(ix), fy0 = floorf(iy);
    const int   x0 = (int)fx0, y0 = (int)fy0;
    float wx1 = ix - fx0, wy1 = iy - fy0;
    float wx0 = 1.0f - wx1, wy0 = 1.0f - wy1;
    // padding_mode='zeros': fold per-corner validity into the weights
    wx0 *= (x0     >= 0 && x0     < W) ? 1.0f : 0.0f;
    wx1 *= (x0 + 1 >= 0 && x0 + 1 < W) ? 1.0f : 0.0f;
    wy0 *= (y0     >= 0 && y0     < H) ? 1.0f : 0.0f;
    wy1 *= (y0 + 1 >= 0 && y0 + 1 < H) ? 1.0f : 0.0f;
    const int xc0 = min(max(x0, 0), W - 1);
    const int xc1 = min(max(x0 + 1, 0), W - 1);
    const int yc0 = min(max(y0, 0), H - 1);
    const int yc1 = min(max(y0 + 1, 0), H - 1);
    off00[j] = yc0 * W + xc0;  off01[j] = yc0 * W + xc1;
    off10[j] = yc1 * W + xc0;  off11[j] = yc1 * W + xc1;
    q00[j] = wx0 * wy0;  q01[j] = wx1 * wy0;
    q10[j] = wx0 * wy1;  q11[j] = wx1 * wy1;
  }

#pragma unroll 4
  for (int c = 0; c < C; ++c) {
    const float* yp = y + (size_t)c * (H * (size_t)W);   // y stays in 192MB L2
    v4f r;
#pragma unroll
    for (int j = 0; j < 4; ++j) {
      r[j] = yp[off00[j]] * q00[j] + yp[off01[j]] * q01[j]
           + yp[off10[j]] * q10[j] + yp[off11[j]] * q11[j];
    }
    float* o = out + ((((size_t)(C + c)) * ND + d) * H + h) * (size_t)W + w;
    __builtin_nontemporal_store(r, (v4f*)o);             // don't thrash L2
  }
}

extern "C" void kernel_launch(void* const* d_in, const int* in_sizes, int n_in,
                              void* d_out, int out_size, void* d_ws, size_t ws_size,
                              hipStream_t stream) {
  (void)in_sizes; (void)n_in; (void)d_ws; (void)ws_size; (void)out_size;
  const float* x     = (const float*)d_in[0];
  const float* y     = (const float*)d_in[1];
  const float* depth = (const float*)d_in[2];
  float* out = (float*)d_out;

  dim3 blk(256);
  // channels 0..15: broadcast copy (16*256*128 / 256 = 2048 blocks)
  cv_ref_broadcast<<<dim3((C * H * (W / 4)) / 256), blk, 0, stream>>>(x, out);
  // channels 16..31: warp (32*256*128 / 256 = 4096 blocks)
  cv_warp<<<dim3((ND * H * (W / 4)) / 256), blk, 0, stream>>>(y, depth, out);
}